// BaseHomogenousModel_80985903334180
// MI455X (gfx1250) — compile-verified
//
#include <hip/hip_runtime.h>

typedef __attribute__((ext_vector_type(2))) float v2f;
typedef __attribute__((ext_vector_type(8))) float v8f;

#define D 128
#define HEADS 2
#define CCH 64
#define TT 22
#define SLOPE 0.2f

// ---------------------------------------------------------------------------
// GEMM: hW = h @ W   ([N,128] x [128,128]) using V_WMMA_F32_16X16X4_F32.
// One wave computes one 16x16 output tile; K-loop of 32 steps (k=4 each).
// f32 WMMA keeps exact reference precision; GEMM is tiny vs. edge traffic.
// ---------------------------------------------------------------------------
__global__ __launch_bounds__(128) void k_gemm_wmma(const float* __restrict__ h,
                                                   const float* __restrict__ W,
                                                   float* __restrict__ hW,
                                                   int n_rows) {
    int wave = blockIdx.x * (blockDim.x >> 5) + (threadIdx.x >> 5);
    int lane = threadIdx.x & 31;
    int mtile = wave >> 3;      // N/16 tiles in M
    int ntile = wave & 7;       // 128/16 = 8 tiles in N
    if (mtile * 16 >= n_rows) return;   // whole-wave uniform (EXEC stays all-ones)

    int halfLane = lane & 15;
    int kphase   = (lane >> 4) << 1;    // lanes 0-15 -> K%4 in {0,1}; 16-31 -> {2,3}
    const float* hrow = h + (size_t)(mtile * 16 + halfLane) * D;
    int col = ntile * 16 + halfLane;

    v8f acc = {};
    for (int kk = 0; kk < D; kk += 4) {
        v2f a, b;
        a.x = hrow[kk + kphase];
        a.y = hrow[kk + kphase + 1];
        b.x = W[(kk + kphase) * D + col];
        b.y = W[(kk + kphase + 1) * D + col];
        acc = __builtin_amdgcn_wmma_f32_16x16x4_f32(false, a, false, b,
                                                    (short)0, acc, false, false);
    }
    // C/D layout: VGPR r -> lanes 0-15: M=r ; lanes 16-31: M=8+r ; N=lane%16
    int baseM = mtile * 16 + ((lane >> 4) << 3);
#pragma unroll
    for (int r = 0; r < 8; ++r)
        hW[(size_t)(baseM + r) * D + col] = acc[r];
}

// ---------------------------------------------------------------------------
// Attention coefficients: a_s[n,h] = dot(hW[n,h,:], asrc[h,:]) (64-wide),
// one wave32 per (node, head); shfl_xor tree reduction.
// ---------------------------------------------------------------------------
__global__ __launch_bounds__(256) void k_attn_coef(const float* __restrict__ hW,
                                                   const float* __restrict__ asrc,
                                                   const float* __restrict__ adst,
                                                   float* __restrict__ a_s,
                                                   float* __restrict__ a_d,
                                                   int n_nodes) {
    int wave = blockIdx.x * 8 + (threadIdx.x >> 5);
    int lane = threadIdx.x & 31;
    if (wave >= n_nodes * HEADS) return;
    int node = wave >> 1, head = wave & 1;
    const float* hp = hW + (size_t)node * D + head * CCH;
    const float* as = asrc + head * CCH;
    const float* ad = adst + head * CCH;
    float h0 = hp[lane], h1 = hp[lane + 32];
    float vs = h0 * as[lane] + h1 * as[lane + 32];
    float vd = h0 * ad[lane] + h1 * ad[lane + 32];
#pragma unroll
    for (int off = 16; off > 0; off >>= 1) {
        vs += __shfl_xor(vs, off, 32);
        vd += __shfl_xor(vd, off, 32);
    }
    if (lane == 0) { a_s[wave] = vs; a_d[wave] = vd; }
}

__global__ void k_init_ms(float* __restrict__ m, float* __restrict__ s, int n) {
    int i = blockIdx.x * blockDim.x + threadIdx.x;
    if (i < n) { m[i] = -__builtin_inff(); s[i] = 0.f; }
}

__global__ void k_init_out(float* __restrict__ out, const float* __restrict__ bias,
                           int total) {
    int i = blockIdx.x * blockDim.x + threadIdx.x;
    if (i < total) out[i] = bias[i & (D - 1)];
}

// Monotonic float atomic max via int/uint punning -> global_atomic_max_i32/min_u32
__device__ __forceinline__ void atomicMaxF(float* addr, float v) {
    if (v >= 0.f) atomicMax((int*)addr, __float_as_int(v));
    else          atomicMin((unsigned int*)addr, __float_as_uint(v));
}

__device__ __forceinline__ void loadEdge(const int* __restrict__ ei, int e, int E,
                                         int& src, int& dst) {
    if (e < E) { src = ei[e]; dst = ei[E + e]; }
    else       { src = e - E; dst = e - E; }       // self loops
}

__global__ __launch_bounds__(256) void k_edge_max(const int* __restrict__ ei, int E,
                                                  int n_nodes,
                                                  const float* __restrict__ a_s,
                                                  const float* __restrict__ a_d,
                                                  float* __restrict__ m) {
    int e = blockIdx.x * blockDim.x + threadIdx.x;
    if (e >= E + n_nodes) return;
    int src, dst; loadEdge(ei, e, E, src, dst);
#pragma unroll
    for (int h = 0; h < HEADS; ++h) {
        float ev = a_s[src * HEADS + h] + a_d[dst * HEADS + h];
        ev = ev > 0.f ? ev : SLOPE * ev;
        atomicMaxF(&m[dst * HEADS + h], ev);
    }
}

__global__ __launch_bounds__(256) void k_edge_sum(const int* __restrict__ ei, int E,
                                                  int n_nodes,
                                                  const float* __restrict__ a_s,
                                                  const float* __restrict__ a_d,
                                                  const float* __restrict__ m,
                                                  float* __restrict__ s) {
    int e = blockIdx.x * blockDim.x + threadIdx.x;
    if (e >= E + n_nodes) return;
    int src, dst; loadEdge(ei, e, E, src, dst);
#pragma unroll
    for (int h = 0; h < HEADS; ++h) {
        float ev = a_s[src * HEADS + h] + a_d[dst * HEADS + h];
        ev = ev > 0.f ? ev : SLOPE * ev;
        unsafeAtomicAdd(&s[dst * HEADS + h], __expf(ev - m[dst * HEADS + h]));
    }
}

// 128 threads per edge: lane c accumulates hW[src,c]*alpha into out[dst,c]
__global__ __launch_bounds__(256) void k_edge_aggr(const int* __restrict__ ei, int E,
                                                   int n_nodes,
                                                   const float* __restrict__ a_s,
                                                   const float* __restrict__ a_d,
                                                   const float* __restrict__ m,
                                                   const float* __restrict__ s,
                                                   const float* __restrict__ hW,
                                                   float* __restrict__ out) {
    unsigned int gid = blockIdx.x * 256u + threadIdx.x;
    int e = (int)(gid >> 7);
    int c = (int)(gid & 127u);
    if (e >= E + n_nodes) return;
    int src, dst; loadEdge(ei, e, E, src, dst);
    int h = c >> 6;
    float ev = a_s[src * HEADS + h] + a_d[dst * HEADS + h];
    ev = ev > 0.f ? ev : SLOPE * ev;
    float alpha = __expf(ev - m[dst * HEADS + h]) / (s[dst * HEADS + h] + 1e-16f);
    unsafeAtomicAdd(&out[(size_t)dst * D + c], hW[(size_t)src * D + c] * alpha);
}

__global__ void k_relu(float* __restrict__ v, int n) {
    int i = blockIdx.x * blockDim.x + threadIdx.x;
    if (i < n) v[i] = fmaxf(v[i], 0.f);
}

// ---------------------------------------------------------------------------
// zf = relu(concat(h[node0], dtype_vec) @ shared_W + shared_b)  -> [256, 64]
// ---------------------------------------------------------------------------
__global__ __launch_bounds__(64) void k_zf(const float* __restrict__ hfin,
                                           const float* __restrict__ x,
                                           const float* __restrict__ sW,
                                           const float* __restrict__ sb,
                                           float* __restrict__ zf, int npg) {
    int g = blockIdx.x, t = threadIdx.x;
    __shared__ float in[D + TT];
    const float* hrow = hfin + (size_t)(g * npg) * D;
    const float* xrow = x + (size_t)(g * npg) * D;
    for (int i = t; i < D; i += 64) in[i] = hrow[i];
    if (t < TT) in[D + t] = xrow[t];
    __syncthreads();
    float acc = sb[t];
    for (int i = 0; i < D + TT; ++i) acc += in[i] * sW[i * 64 + t];
    zf[g * 64 + t] = fmaxf(acc, 0.f);
}

// ---------------------------------------------------------------------------
// All MLP heads, one block per graph.  s_in layout:
//   [0,64)=zf  [64,86)=dtype  [86,91)=cat  [91]=signed [92]=floating [93]=bool
// ---------------------------------------------------------------------------
__global__ __launch_bounds__(64) void k_heads(
    const float* __restrict__ zf, const float* __restrict__ x, int npg,
    const float* ptr_W1, const float* ptr_b1, const float* ptr_W2, const float* ptr_b2,
    const float* cat_W1, const float* cat_b1, const float* cat_W2, const float* cat_b2,
    const float* sf_W1,  const float* sf_b1,  const float* sf_W2,  const float* sf_b2,
    const float* bool_W1,const float* bool_b1,const float* bool_W2,const float* bool_b2,
    const float* size_W1,const float* size_b1,const float* size_W2,const float* size_b2,
    float* __restrict__ out, int NG) {
    int g = blockIdx.x, t = threadIdx.x;
    __shared__ float s_in[94];
    __shared__ float s_h[64];
    s_in[t] = zf[g * 64 + t];
    if (t < TT) s_in[64 + t] = x[(size_t)(g * npg) * D + t];
    __syncthreads();

    // ptr heads (3x): 64 -> 64 relu -> 3
    for (int k = 0; k < 3; ++k) {
        float a = ptr_b1[k * 64 + t];
        for (int i = 0; i < 64; ++i) a += s_in[i] * ptr_W1[k * 4096 + i * 64 + t];
        s_h[t] = fmaxf(a, 0.f);
        __syncthreads();
        if (t < 3) {
            float o = ptr_b2[k * 3 + t];
            for (int i = 0; i < 64; ++i) o += s_h[i] * ptr_W2[k * 192 + i * 3 + t];
            out[k * (NG * 3) + g * 3 + t] = o;
        }
        __syncthreads();
    }
    // cat: 64 -> 64 relu -> 5
    {
        float a = cat_b1[t];
        for (int i = 0; i < 64; ++i) a += s_in[i] * cat_W1[i * 64 + t];
        s_h[t] = fmaxf(a, 0.f);
        __syncthreads();
        if (t < 5) {
            float o = cat_b2[t];
            for (int i = 0; i < 64; ++i) o += s_h[i] * cat_W2[i * 5 + t];
            s_in[86 + t] = o;
            out[3 * NG * 3 + g * 5 + t] = o;
        }
        __syncthreads();
    }
    // sf (signed / floating): 64 -> 64 relu -> 1
    for (int k = 0; k < 2; ++k) {
        float a = sf_b1[k * 64 + t];
        for (int i = 0; i < 64; ++i) a += s_in[i] * sf_W1[k * 4096 + i * 64 + t];
        s_h[t] = fmaxf(a, 0.f);
        __syncthreads();
        if (t == 0) {
            float o = sf_b2[k];
            for (int i = 0; i < 64; ++i) o += s_h[i] * sf_W2[k * 64 + i];
            s_in[91 + k] = o;
            out[3 * NG * 3 + NG * 5 + k * NG + g] = o;
        }
        __syncthreads();
    }
    // bool: 93 -> 64 relu -> 1
    {
        float a = bool_b1[t];
        for (int i = 0; i < 93; ++i) a += s_in[i] * bool_W1[i * 64 + t];
        s_h[t] = fmaxf(a, 0.f);
        __syncthreads();
        if (t == 0) {
            float o = bool_b2[0];
            for (int i = 0; i < 64; ++i) o += s_h[i] * bool_W2[i];
            s_in[93] = o;
            out[3 * NG * 3 + NG * 5 + 2 * NG + g] = o;
        }
        __syncthreads();
    }
    // size: 94 -> 64 relu -> 8
    {
        float a = size_b1[t];
        for (int i = 0; i < 94; ++i) a += s_in[i] * size_W1[i * 64 + t];
        s_h[t] = fmaxf(a, 0.f);
        __syncthreads();
        if (t < 8) {
            float o = size_b2[t];
            for (int i = 0; i < 64; ++i) o += s_h[i] * size_W2[i * 8 + t];
            out[3 * NG * 3 + NG * 5 + 3 * NG + g * 8 + t] = o;
        }
    }
}

extern "C" void kernel_launch(void* const* d_in, const int* in_sizes, int n_in,
                              void* d_out, int out_size, void* d_ws, size_t ws_size,
                              hipStream_t stream) {
    const float* x        = (const float*)d_in[0];
    const int*   ei       = (const int*)d_in[1];
    const float* gat_W    = (const float*)d_in[3];
    const float* gat_asrc = (const float*)d_in[4];
    const float* gat_adst = (const float*)d_in[5];
    const float* gat_b    = (const float*)d_in[6];
    const float* shared_W = (const float*)d_in[7];
    const float* shared_b = (const float*)d_in[8];

    int N  = in_sizes[0] / D;
    int E  = in_sizes[1] / 2;
    int NG = 256;
    int NB = in_sizes[2];
    int npg = NB / NG;

    float* ws   = (float*)d_ws;
    size_t nd   = (size_t)N * D;
    float* bufA = ws;
    float* bufB = bufA + nd;
    float* hW   = bufB + nd;
    float* a_s  = hW + nd;
    float* a_d  = a_s + (size_t)N * HEADS;
    float* m    = a_d + (size_t)N * HEADS;
    float* s    = m + (size_t)N * HEADS;
    float* zf   = s + (size_t)N * HEADS;

    const float* h = x;
    float* outbuf[3] = {bufA, bufB, bufA};
    int total = E + N;

    for (int l = 0; l < 3; ++l) {
        float* out = outbuf[l];
        int tiles = (N / 16) * (D / 16);
        k_gemm_wmma<<<(tiles + 3) / 4, 128, 0, stream>>>(h, gat_W + (size_t)l * D * D, hW, N);
        k_attn_coef<<<(N * HEADS + 7) / 8, 256, 0, stream>>>(
            hW, gat_asrc + l * HEADS * CCH, gat_adst + l * HEADS * CCH, a_s, a_d, N);
        k_init_ms<<<(N * HEADS + 255) / 256, 256, 0, stream>>>(m, s, N * HEADS);
        k_init_out<<<(int)((nd + 255) / 256), 256, 0, stream>>>(out, gat_b + l * D, (int)nd);
        k_edge_max<<<(total + 255) / 256, 256, 0, stream>>>(ei, E, N, a_s, a_d, m);
        k_edge_sum<<<(total + 255) / 256, 256, 0, stream>>>(ei, E, N, a_s, a_d, m, s);
        unsigned int aggrBlocks = (unsigned int)(((long long)total * D + 255) / 256);
        k_edge_aggr<<<aggrBlocks, 256, 0, stream>>>(ei, E, N, a_s, a_d, m, s, hW, out);
        if (l < 2) k_relu<<<(int)((nd + 255) / 256), 256, 0, stream>>>(out, (int)nd);
        h = out;
    }

    k_zf<<<NG, 64, 0, stream>>>(h, x, shared_W, shared_b, zf, npg);
    k_heads<<<NG, 64, 0, stream>>>(zf, x, npg,
        (const float*)d_in[9],  (const float*)d_in[10], (const float*)d_in[11], (const float*)d_in[12],
        (const float*)d_in[13], (const float*)d_in[14], (const float*)d_in[15], (const float*)d_in[16],
        (const float*)d_in[17], (const float*)d_in[18], (const float*)d_in[19], (const float*)d_in[20],
        (const float*)d_in[21], (const float*)d_in[22], (const float*)d_in[23], (const float*)d_in[24],
        (const float*)d_in[25], (const float*)d_in[26], (const float*)d_in[27], (const float*)d_in[28],
        (float*)d_out, NG);
}